// ForwardDeformer_86577950753310
// MI455X (gfx1250) — compile-verified
//
#include <hip/hip_runtime.h>
#include <hip/hip_bf16.h>
#include <math.h>

// ---------------------------------------------------------------- constants
#define RES        128
#define ZRATIO     4
#define DD         (RES / ZRATIO)   // 32
#define HH         RES              // 128
#define WW         RES              // 128
#define VV         (DD * HH * WW)   // 524288
#define NJ         24
#define NB         2
#define NN         65536
#define NI         9
#define SOFT_BLEND 20.0f
#define CVG        1e-4f
#define DVG        1.0f
#define NITERS     10
#define DUP_EPS2   1e-6f

typedef float v2f __attribute__((ext_vector_type(2)));
typedef float v8f __attribute__((ext_vector_type(8)));

__constant__ int c_init_bones[NI] = {0, 1, 2, 4, 5, 16, 17, 18, 19};

// ---------------------------------------------------------------- helpers
__device__ __forceinline__ void inv3x3(const float m[9], float r[9]) {
    // mirrors reference inv3() exactly (adjugate / clamped det)
    float a = m[0], b = m[1], c = m[2];
    float d = m[3], e = m[4], f = m[5];
    float g = m[6], h = m[7], i = m[8];
    float A  = e * i - f * h;
    float B_ = c * h - b * i;
    float C  = b * f - c * e;
    float Dd = f * g - d * i;
    float E  = a * i - c * g;
    float F  = c * d - a * f;
    float G  = d * h - e * g;
    float Hh = b * g - a * h;
    float Ii = a * e - b * d;
    float det = a * A + b * Dd + c * G;
    if (fabsf(det) < 1e-12f) det = 1e-12f;
    float id = 1.0f / det;
    r[0] = A * id;  r[1] = B_ * id; r[2] = C * id;
    r[3] = Dd * id; r[4] = E * id;  r[5] = F * id;
    r[6] = G * id;  r[7] = Hh * id; r[8] = Ii * id;
}

// trilinear grid_sample_3d, align_corners=True, border-clamped index +
// zeroed weight outside (matches reference semantics exactly).
template <int C>
__device__ __forceinline__ void sample3d(const float* __restrict__ vol,
                                         float cx, float cy, float cz,
                                         float out[C]) {
    float ix = (cx + 1.0f) * 0.5f * (float)(WW - 1);
    float iy = (cy + 1.0f) * 0.5f * (float)(HH - 1);
    float iz = (cz + 1.0f) * 0.5f * (float)(DD - 1);
    float xf = floorf(ix), yf = floorf(iy), zf = floorf(iz);
    float tx = ix - xf, ty = iy - yf, tz = iz - zf;
    int x0 = (int)xf, y0 = (int)yf, z0 = (int)zf;
#pragma unroll
    for (int c = 0; c < C; ++c) out[c] = 0.0f;
#pragma unroll
    for (int dz = 0; dz < 2; ++dz)
#pragma unroll
        for (int dy = 0; dy < 2; ++dy)
#pragma unroll
            for (int dx = 0; dx < 2; ++dx) {
                int xi = x0 + dx, yi = y0 + dy, zi = z0 + dz;
                bool inb = (xi >= 0) && (xi < WW) && (yi >= 0) && (yi < HH) &&
                           (zi >= 0) && (zi < DD);
                float wgt = (dx ? tx : 1.0f - tx) * (dy ? ty : 1.0f - ty) *
                            (dz ? tz : 1.0f - tz);
                wgt = inb ? wgt : 0.0f;
                int xc = min(max(xi, 0), WW - 1);
                int yc = min(max(yi, 0), HH - 1);
                int zc = min(max(zi, 0), DD - 1);
                size_t lin = ((size_t)zc * HH + yc) * WW + xc;
#pragma unroll
                for (int c = 0; c < C; ++c)
                    out[c] += vol[(size_t)c * VV + lin] * wgt;
            }
}

// ---------------------------------------------------------------- K1: softmax over joints
__global__ void softmax_kernel(const float* __restrict__ lbs,
                               float* __restrict__ w) {
    int v = blockIdx.x * blockDim.x + threadIdx.x;  // v < VV (exact grid)
    float vals[NJ];
    float m = -1e30f;
#pragma unroll
    for (int j = 0; j < NJ; ++j) {
        vals[j] = SOFT_BLEND * lbs[(size_t)j * VV + v];
        m = fmaxf(m, vals[j]);
    }
    float s = 0.0f;
#pragma unroll
    for (int j = 0; j < NJ; ++j) {
        vals[j] = expf(vals[j] - m);
        s += vals[j];
    }
    float inv = 1.0f / s;
#pragma unroll
    for (int j = 0; j < NJ; ++j) w[(size_t)j * VV + v] = vals[j] * inv;
}

// ---------------------------------------------------------------- K2: WMMA blend + inv3
// Per wave: 32 voxels as two 16x16 WMMA tiles.
//   GEMM: D[m(voxel), n(T-entry)] = sum_k w[k][v]*tfs[b][k][n],  K=24 in 6 steps of 4.
// A 16x4 f32 layout (ISA): lanes0-15 M=lane {VGPR0=K0, VGPR1=K1},
//                          lanes16-31 M=lane-16 {VGPR0=K2, VGPR1=K3}; B mirrored.
// D layout: VGPR r = row (r | r+8 by lane half), lane&15 = column n.
__global__ void precompute_kernel(const float* __restrict__ w,
                                  const float* __restrict__ tfs,
                                  const float* __restrict__ offset,
                                  const float* __restrict__ scale,
                                  float* __restrict__ vox_d,
                                  float* __restrict__ vox_J) {
    __shared__ float lds[8][32][16];
    const int lane = threadIdx.x & 31;
    const int wv   = threadIdx.x >> 5;
    const int blocksPerBatch = VV / 256;
    const int bb   = blockIdx.x / blocksPerBatch;
    const int vblk = (blockIdx.x % blocksPerBatch) * 256 + wv * 32;
    const int half = (lane < 16) ? 0 : 1;
    const int nl   = lane & 15;

#pragma unroll
    for (int t = 0; t < 2; ++t) {
        const int vbase = vblk + t * 16;
        v8f acc = {};
#pragma unroll
        for (int kk = 0; kk < 6; ++kk) {
            const int k0 = kk * 4 + half * 2;   // rows (k0, k0+1) for this lane half
            v2f a, bm;
            const int vA = vbase + nl;          // voxel for M=nl
            a.x = w[(size_t)(k0 + 0) * VV + vA];
            a.y = w[(size_t)(k0 + 1) * VV + vA];
            const int ncl = min(nl, 11);        // in-range load, select below
            float b0 = tfs[((size_t)bb * NJ + (k0 + 0)) * 16 + ncl];
            float b1 = tfs[((size_t)bb * NJ + (k0 + 1)) * 16 + ncl];
            bm.x = (nl < 12) ? b0 : 0.0f;
            bm.y = (nl < 12) ? b1 : 0.0f;
            acc = __builtin_amdgcn_wmma_f32_16x16x4_f32(
                /*neg_a=*/false, a, /*neg_b=*/false, bm,
                /*c_mod=*/(short)0, acc, /*reuse_a=*/false, /*reuse_b=*/false);
        }
#pragma unroll
        for (int r = 0; r < 8; ++r)
            lds[wv][t * 16 + r + half * 8][nl] = acc[r];
    }
    __syncthreads();

    // epilogue: each lane owns one of the wave's 32 voxels
    const int v = vblk + lane;
    float T[12];
#pragma unroll
    for (int e = 0; e < 12; ++e) T[e] = lds[wv][lane][e];

    const int iz  = v / (HH * WW);
    const int rem = v % (HH * WW);
    const int iy  = rem / WW;
    const int ixx = rem % WW;
    const float gx = -1.0f + 2.0f * (float)ixx / (float)(WW - 1);
    const float gy = -1.0f + 2.0f * (float)iy  / (float)(HH - 1);
    const float gz = -1.0f + 2.0f * (float)iz  / (float)(DD - 1);
    const float g0 = gx / scale[0] - offset[0];
    const float g1 = gy / scale[1] - offset[1];
    const float g2 = gz / scale[2] - offset[2];

#pragma unroll
    for (int r = 0; r < 3; ++r) {
        float dr = T[r * 4 + 0] * g0 + T[r * 4 + 1] * g1 + T[r * 4 + 2] * g2 +
                   T[r * 4 + 3];
        vox_d[((size_t)bb * 3 + r) * VV + v] = dr;
    }
    float M[9] = {T[0], T[1], T[2], T[4], T[5], T[6], T[8], T[9], T[10]};
    float Ji[9];
    inv3x3(M, Ji);
#pragma unroll
    for (int k = 0; k < 9; ++k)
        vox_J[((size_t)bb * 9 + k) * VV + v] = Ji[k];
}

// ---------------------------------------------------------------- K3: Broyden (1 thread / point)
__global__ void broyden_kernel(const float* __restrict__ xd,
                               const float* __restrict__ tfs,
                               const float* __restrict__ vox_d,
                               const float* __restrict__ vox_J,
                               const float* __restrict__ offset,
                               const float* __restrict__ scale,
                               float* __restrict__ out_xc,
                               float* __restrict__ out_norm) {
    const int p = blockIdx.x * blockDim.x + threadIdx.x;  // < NB*NN*NI (exact)
    const int b   = p / (NN * NI);
    const int rem = p % (NN * NI);
    const int n   = rem / NI;
    const int i   = rem % NI;
    const int bone = c_init_bones[i];

    const float* tf = tfs + ((size_t)b * NJ + bone) * 16;
    float A[9] = {tf[0], tf[1], tf[2], tf[4], tf[5], tf[6], tf[8], tf[9], tf[10]};
    const float t0 = tf[3], t1 = tf[7], t2 = tf[11];
    float Ainv[9];
    inv3x3(A, Ainv);

    const float xd0 = xd[((size_t)b * NN + n) * 3 + 0];
    const float xd1 = xd[((size_t)b * NN + n) * 3 + 1];
    const float xd2 = xd[((size_t)b * NN + n) * 3 + 2];
    const float r0 = xd0 - t0, r1 = xd1 - t1, r2 = xd2 - t2;

    float x[3];
    x[0] = Ainv[0] * r0 + Ainv[1] * r1 + Ainv[2] * r2;
    x[1] = Ainv[3] * r0 + Ainv[4] * r1 + Ainv[5] * r2;
    x[2] = Ainv[6] * r0 + Ainv[7] * r1 + Ainv[8] * r2;

    const float of0 = offset[0], of1 = offset[1], of2 = offset[2];
    const float sc0 = scale[0], sc1 = scale[1], sc2 = scale[2];
    const float* vd = vox_d + (size_t)b * 3 * VV;
    const float* vj = vox_J + (size_t)b * 9 * VV;

    float Jinv[9];
    sample3d<9>(vj, (x[0] + of0) * sc0, (x[1] + of1) * sc1, (x[2] + of2) * sc2,
                Jinv);

    float g[3];
    {
        float d[3];
        sample3d<3>(vd, (x[0] + of0) * sc0, (x[1] + of1) * sc1,
                    (x[2] + of2) * sc2, d);
        g[0] = d[0] - xd0; g[1] = d[1] - xd1; g[2] = d[2] - xd2;
    }
    float gn = sqrtf(g[0] * g[0] + g[1] * g[1] + g[2] * g[2]);
    float bx0 = x[0], bx1 = x[1], bx2 = x[2];
    float bn = gn;
    bool active = (gn >= CVG) && (gn <= DVG);

    for (int it = 0; it < NITERS; ++it) {
        if (active) {
            float dx[3];
            dx[0] = -(Jinv[0] * g[0] + Jinv[1] * g[1] + Jinv[2] * g[2]);
            dx[1] = -(Jinv[3] * g[0] + Jinv[4] * g[1] + Jinv[5] * g[2]);
            dx[2] = -(Jinv[6] * g[0] + Jinv[7] * g[1] + Jinv[8] * g[2]);
            x[0] += dx[0]; x[1] += dx[1]; x[2] += dx[2];

            float gnew[3];
            {
                float d[3];
                sample3d<3>(vd, (x[0] + of0) * sc0, (x[1] + of1) * sc1,
                            (x[2] + of2) * sc2, d);
                gnew[0] = d[0] - xd0; gnew[1] = d[1] - xd1; gnew[2] = d[2] - xd2;
            }
            float dg[3] = {gnew[0] - g[0], gnew[1] - g[1], gnew[2] - g[2]};
            float Jdg[3];
            Jdg[0] = Jinv[0] * dg[0] + Jinv[1] * dg[1] + Jinv[2] * dg[2];
            Jdg[1] = Jinv[3] * dg[0] + Jinv[4] * dg[1] + Jinv[5] * dg[2];
            Jdg[2] = Jinv[6] * dg[0] + Jinv[7] * dg[1] + Jinv[8] * dg[2];
            float denom = dx[0] * Jdg[0] + dx[1] * Jdg[1] + dx[2] * Jdg[2];
            if (fabsf(denom) < 1e-12f) denom = 1e-12f;
            const float idn = 1.0f / denom;
            float u[3] = {(dx[0] - Jdg[0]) * idn, (dx[1] - Jdg[1]) * idn,
                          (dx[2] - Jdg[2]) * idn};
            float vT[3];
            vT[0] = dx[0] * Jinv[0] + dx[1] * Jinv[3] + dx[2] * Jinv[6];
            vT[1] = dx[0] * Jinv[1] + dx[1] * Jinv[4] + dx[2] * Jinv[7];
            vT[2] = dx[0] * Jinv[2] + dx[1] * Jinv[5] + dx[2] * Jinv[8];
#pragma unroll
            for (int rr = 0; rr < 3; ++rr)
#pragma unroll
                for (int cc = 0; cc < 3; ++cc)
                    Jinv[rr * 3 + cc] += u[rr] * vT[cc];

            g[0] = gnew[0]; g[1] = gnew[1]; g[2] = gnew[2];
            float gn2 = sqrtf(g[0] * g[0] + g[1] * g[1] + g[2] * g[2]);
            if (gn2 < bn) { bn = gn2; bx0 = x[0]; bx1 = x[1]; bx2 = x[2]; }
            active = (gn2 >= CVG) && (gn2 <= DVG);
        }
    }
    // correction = xd_opt - stop_gradient(xd_opt) == 0 in a value-only forward
    // pass, so the final xc equals xc_opt exactly (forward_skinning elided).
    out_xc[(size_t)p * 3 + 0] = bx0;
    out_xc[(size_t)p * 3 + 1] = bx1;
    out_xc[(size_t)p * 3 + 2] = bx2;
    out_norm[p] = bn;
}

// ---------------------------------------------------------------- K4: duplicate filter
__global__ void filter_kernel(const float* __restrict__ xc,
                              const float* __restrict__ norms,
                              float* __restrict__ out_valid) {
    const int q = blockIdx.x * blockDim.x + threadIdx.x;  // (b*NN + n), exact
    float px[NI][3];
    bool val[NI];
#pragma unroll
    for (int i = 0; i < NI; ++i) {
        const size_t base = ((size_t)q * NI + i) * 3;
        px[i][0] = xc[base + 0];
        px[i][1] = xc[base + 1];
        px[i][2] = xc[base + 2];
        val[i] = norms[(size_t)q * NI + i] < CVG;
    }
#pragma unroll
    for (int i = 0; i < NI; ++i) {
        bool dup = false;
#pragma unroll
        for (int j = 0; j < NI; ++j) {
            if (j < i) {
                float d0 = px[i][0] - px[j][0];
                float d1 = px[i][1] - px[j][1];
                float d2 = px[i][2] - px[j][2];
                float dd = d0 * d0 + d1 * d1 + d2 * d2;
                dup = dup || (val[j] && (dd < DUP_EPS2));
            }
        }
        out_valid[(size_t)q * NI + i] = (val[i] && !dup) ? 1.0f : 0.0f;
    }
}

// ---------------------------------------------------------------- launch
extern "C" void kernel_launch(void* const* d_in, const int* in_sizes, int n_in,
                              void* d_out, int out_size, void* d_ws,
                              size_t ws_size, hipStream_t stream) {
    (void)in_sizes; (void)n_in; (void)out_size; (void)ws_size;
    const float* xd     = (const float*)d_in[0];  // (B,N,3)
    const float* tfs    = (const float*)d_in[1];  // (B,24,4,4)
    const float* lbs    = (const float*)d_in[2];  // (1,24,D,H,W)
    const float* offset = (const float*)d_in[3];  // (1,1,3)
    const float* scale  = (const float*)d_in[4];  // (1,1,3)
    float* out = (float*)d_out;
    float* ws  = (float*)d_ws;

    // workspace layout (floats): weights | voxel_d | voxel_J | best_norm
    float* w      = ws;                               // NJ*VV        (50.3 MB)
    float* vox_d  = w + (size_t)NJ * VV;              // NB*3*VV      (12.6 MB)
    float* vox_J  = vox_d + (size_t)NB * 3 * VV;      // NB*9*VV      (37.7 MB)
    float* norms  = vox_J + (size_t)NB * 9 * VV;      // NB*NN*NI     ( 4.7 MB)

    float* out_xc    = out;                           // NB*NN*NI*3
    float* out_valid = out + (size_t)NB * NN * NI * 3;// NB*NN*NI

    softmax_kernel<<<VV / 256, 256, 0, stream>>>(lbs, w);
    precompute_kernel<<<NB * (VV / 256), 256, 0, stream>>>(w, tfs, offset,
                                                           scale, vox_d, vox_J);
    broyden_kernel<<<(NB * NN * NI) / 256, 256, 0, stream>>>(
        xd, tfs, vox_d, vox_J, offset, scale, out_xc, norms);
    filter_kernel<<<(NB * NN) / 256, 256, 0, stream>>>(out_xc, norms,
                                                       out_valid);
}